// GENConv_22170621182407
// MI455X (gfx1250) — compile-verified
//
#include <hip/hip_runtime.h>
#include <math.h>

#define DD 128
#define EPS_F 1e-7f

typedef __attribute__((ext_vector_type(2))) float v2f;
typedef __attribute__((ext_vector_type(8))) float v8f;

// ---------------------------------------------------------------------------
// Kernel 0: zero the atomic accumulators (segmax, sum_ex, sum_mex)
// ---------------------------------------------------------------------------
__global__ void zero_kernel(float4* __restrict__ p, long long n4) {
    long long i = (long long)blockIdx.x * blockDim.x + threadIdx.x;
    if (i < n4) p[i] = make_float4(0.f, 0.f, 0.f, 0.f);
}

// ---------------------------------------------------------------------------
// Kernel 1: per-edge m = relu(nf[src]+ef)+eps ; atomic segment-max into
// segmax[dst]. All m > 0, so integer atomicMax on the float bit pattern is
// order-correct and the 0-init acts as -inf for non-empty segments.
// One wave (32 lanes) per edge, float4 per lane -> 512B coalesced rows.
// ---------------------------------------------------------------------------
__global__ void edge_pass1(const float* __restrict__ nf, const float* __restrict__ ef,
                           const int* __restrict__ src, const int* __restrict__ dst,
                           int* __restrict__ segmax_i, int E) {
    int e = blockIdx.x * 8 + (threadIdx.x >> 5);
    if (e >= E) return;
    int lane = threadIdx.x & 31;
    int s = src[e], d = dst[e];
    float4 a = *(const float4*)(nf + (size_t)s * DD + lane * 4);
    float4 b = *(const float4*)(ef + (size_t)e * DD + lane * 4);
    float m0 = fmaxf(a.x + b.x, 0.f) + EPS_F;
    float m1 = fmaxf(a.y + b.y, 0.f) + EPS_F;
    float m2 = fmaxf(a.z + b.z, 0.f) + EPS_F;
    float m3 = fmaxf(a.w + b.w, 0.f) + EPS_F;
    int* sm = segmax_i + (size_t)d * DD + lane * 4;
    atomicMax(sm + 0, __float_as_int(m0));
    atomicMax(sm + 1, __float_as_int(m1));
    atomicMax(sm + 2, __float_as_int(m2));
    atomicMax(sm + 3, __float_as_int(m3));
}

// ---------------------------------------------------------------------------
// Kernel 2: recompute m, ex = exp(m - segmax[dst]); atomic-add ex and m*ex.
// agg will later be (sum m*ex)/(sum ex) == sum m*softmax(m).
// ---------------------------------------------------------------------------
__global__ void edge_pass2(const float* __restrict__ nf, const float* __restrict__ ef,
                           const int* __restrict__ src, const int* __restrict__ dst,
                           const float* __restrict__ segmax,
                           float* __restrict__ sum_ex, float* __restrict__ sum_mex, int E) {
    int e = blockIdx.x * 8 + (threadIdx.x >> 5);
    if (e >= E) return;
    int lane = threadIdx.x & 31;
    int s = src[e], d = dst[e];
    float4 a = *(const float4*)(nf + (size_t)s * DD + lane * 4);
    float4 b = *(const float4*)(ef + (size_t)e * DD + lane * 4);
    float4 mx = *(const float4*)(segmax + (size_t)d * DD + lane * 4);
    float m0 = fmaxf(a.x + b.x, 0.f) + EPS_F;
    float m1 = fmaxf(a.y + b.y, 0.f) + EPS_F;
    float m2 = fmaxf(a.z + b.z, 0.f) + EPS_F;
    float m3 = fmaxf(a.w + b.w, 0.f) + EPS_F;
    float e0 = __expf(m0 - mx.x);
    float e1 = __expf(m1 - mx.y);
    float e2 = __expf(m2 - mx.z);
    float e3 = __expf(m3 - mx.w);
    float* se  = sum_ex  + (size_t)d * DD + lane * 4;
    float* sme = sum_mex + (size_t)d * DD + lane * 4;
    atomicAdd(se  + 0, e0);       atomicAdd(se  + 1, e1);
    atomicAdd(se  + 2, e2);       atomicAdd(se  + 3, e3);
    atomicAdd(sme + 0, m0 * e0);  atomicAdd(sme + 1, m1 * e1);
    atomicAdd(sme + 2, m2 * e2);  atomicAdd(sme + 3, m3 * e3);
}

// ---------------------------------------------------------------------------
// Kernel 3: per-node agg = sum_mex/sum_ex (0 for empty nodes);
// feats = nf + (agg/max(||agg||,1e-12)) * ||nf||. One wave per node,
// wave32 shfl-xor reduction for the two row norms.
// ---------------------------------------------------------------------------
__global__ void node_finalize(const float* __restrict__ nf,
                              const float* __restrict__ sum_ex,
                              const float* __restrict__ sum_mex,
                              float* __restrict__ feats, int N) {
    int n = blockIdx.x * 8 + (threadIdx.x >> 5);
    if (n >= N) return;
    int lane = threadIdx.x & 31;
    size_t base = (size_t)n * DD + lane * 4;
    float4 den = *(const float4*)(sum_ex + base);
    float4 num = *(const float4*)(sum_mex + base);
    float4 h   = *(const float4*)(nf + base);
    float4 agg;
    agg.x = den.x > 0.f ? num.x / den.x : 0.f;
    agg.y = den.y > 0.f ? num.y / den.y : 0.f;
    agg.z = den.z > 0.f ? num.z / den.z : 0.f;
    agg.w = den.w > 0.f ? num.w / den.w : 0.f;
    float pa = agg.x*agg.x + agg.y*agg.y + agg.z*agg.z + agg.w*agg.w;
    float ph = h.x*h.x + h.y*h.y + h.z*h.z + h.w*h.w;
    #pragma unroll
    for (int off = 16; off >= 1; off >>= 1) {
        pa += __shfl_xor(pa, off, 32);
        ph += __shfl_xor(ph, off, 32);
    }
    float scale = sqrtf(ph) / fmaxf(sqrtf(pa), 1e-12f);
    float4 o;
    o.x = h.x + agg.x * scale;
    o.y = h.y + agg.y * scale;
    o.z = h.z + agg.z * scale;
    o.w = h.w + agg.w * scale;
    *(float4*)(feats + base) = o;
}

// ---------------------------------------------------------------------------
// Kernel 4: out = feats @ W + b using V_WMMA_F32_16X16X4_F32 (exact f32).
// Block = 256 threads = 8 waves, computing a 32x128 output strip.
// Wave w owns N-tile [w*16, w*16+16) and TWO M-tiles (rows 0-15 and 16-31 of
// the strip) with independent accumulators that share each B fragment:
// 2 WMMA per B fetch, back-to-back independent WMMAs (no D->A/B hazard).
// A tile (32x128) staged in LDS with padded stride 132 (bank-conflict free
// column reads); W (64KB) is L2/WGP$-resident and read directly.
// A-frag layout per ISA: lanes 0-15 hold K=k..k+1, lanes 16-31 hold K=k+2..k+3
// for rows M=lane&15. C/D layout: VGPR j -> row j (lanes 0-15) / row j+8
// (lanes 16-31), col = n0 + (lane&15).
// Epilogue: single uniform block-level bounds check -> straight-line stores
// for all full blocks; only the one tail block takes the guarded path.
// ---------------------------------------------------------------------------
#define LDSA_STRIDE 132
#define MTILE 32

__global__ void gemm_wmma(const float* __restrict__ feats, const float* __restrict__ w,
                          const float* __restrict__ bias, float* __restrict__ out, int N) {
    __shared__ float ldsA[MTILE * LDSA_STRIDE];
    int tid  = threadIdx.x;
    int wave = tid >> 5;          // 0..7 -> N tile
    int lane = tid & 31;
    int mbase = blockIdx.x * MTILE;

    // Cooperative load of the 32x128 A tile (float4 per slot, coalesced)
    #pragma unroll
    for (int i = tid; i < MTILE * 32; i += 256) {
        int row = i >> 5;              // 0..31
        int col = (i & 31) << 2;       // 0..124 step 4
        int grow = mbase + row;
        float4 v = (grow < N) ? *(const float4*)(feats + (size_t)grow * DD + col)
                              : make_float4(0.f, 0.f, 0.f, 0.f);
        float* p = &ldsA[row * LDSA_STRIDE + col];
        p[0] = v.x; p[1] = v.y; p[2] = v.z; p[3] = v.w;
    }
    __syncthreads();

    int n0   = wave * 16;
    int half = lane >> 4;      // 0: K=k,k+1 ; 1: K=k+2,k+3
    int l15  = lane & 15;
    v8f c0 = {};
    v8f c1 = {};
    #pragma unroll 4
    for (int k = 0; k < DD; k += 4) {
        int ka = k + half * 2;
        v2f a0, a1, b;
        a0.x = ldsA[l15 * LDSA_STRIDE + ka];
        a0.y = ldsA[l15 * LDSA_STRIDE + ka + 1];
        a1.x = ldsA[(16 + l15) * LDSA_STRIDE + ka];
        a1.y = ldsA[(16 + l15) * LDSA_STRIDE + ka + 1];
        b.x  = w[(size_t)ka * DD + n0 + l15];
        b.y  = w[(size_t)(ka + 1) * DD + n0 + l15];
        c0 = __builtin_amdgcn_wmma_f32_16x16x4_f32(false, a0, false, b,
                                                   (short)0, c0, false, false);
        c1 = __builtin_amdgcn_wmma_f32_16x16x4_f32(false, a1, false, b,
                                                   (short)0, c1, false, false);
    }

    float bv   = bias[n0 + l15];
    int   col  = n0 + l15;
    int   rb   = mbase + half * 8;
    if (mbase + MTILE <= N) {
        // fast path: fully in-bounds strip (all blocks except possibly the last)
        #pragma unroll
        for (int j = 0; j < 8; ++j) {
            out[(size_t)(rb + j)      * DD + col] = c0[j] + bv;
            out[(size_t)(rb + 16 + j) * DD + col] = c1[j] + bv;
        }
    } else {
        #pragma unroll
        for (int j = 0; j < 8; ++j) {
            int r0 = rb + j, r1 = rb + 16 + j;
            if (r0 < N) out[(size_t)r0 * DD + col] = c0[j] + bv;
            if (r1 < N) out[(size_t)r1 * DD + col] = c1[j] + bv;
        }
    }
}

// ---------------------------------------------------------------------------
extern "C" void kernel_launch(void* const* d_in, const int* in_sizes, int n_in,
                              void* d_out, int out_size, void* d_ws, size_t ws_size,
                              hipStream_t stream) {
    const float* nf   = (const float*)d_in[0];
    const float* ef   = (const float*)d_in[1];
    const float* w    = (const float*)d_in[2];
    const float* bias = (const float*)d_in[3];
    const int*   src  = (const int*)d_in[4];
    const int*   dst  = (const int*)d_in[5];

    const int D = in_sizes[3];              // 128
    const int N = in_sizes[0] / D;          // 50000
    const int E = in_sizes[4];              // 600000
    (void)n_in; (void)out_size; (void)ws_size;

    float* segmax  = (float*)d_ws;                     // [N*D]
    float* sum_ex  = segmax  + (size_t)N * D;          // [N*D]
    float* sum_mex = sum_ex  + (size_t)N * D;          // [N*D]
    float* feats   = sum_mex + (size_t)N * D;          // [N*D]
    float* out     = (float*)d_out;

    // 0) zero the three accumulator planes
    long long n4 = (long long)3 * N * D / 4;
    zero_kernel<<<(unsigned)((n4 + 255) / 256), 256, 0, stream>>>((float4*)segmax, n4);

    // 1) segment max (int-atomic on positive float bits)
    edge_pass1<<<(E + 7) / 8, 256, 0, stream>>>(nf, ef, src, dst, (int*)segmax, E);

    // 2) segment sums of ex and m*ex
    edge_pass2<<<(E + 7) / 8, 256, 0, stream>>>(nf, ef, src, dst, segmax,
                                                sum_ex, sum_mex, E);

    // 3) MessageNorm -> feats
    node_finalize<<<(N + 7) / 8, 256, 0, stream>>>(nf, sum_ex, sum_mex, feats, N);

    // 4) feats @ W + b via f32 WMMA (2 M-tiles per wave)
    gemm_wmma<<<(N + MTILE - 1) / MTILE, 256, 0, stream>>>(feats, w, bias, out, N);
}